// Attention_64209761075491
// MI455X (gfx1250) — compile-verified
//
#include <hip/hip_runtime.h>
#include <hip/hip_bf16.h>
#include <math.h>
#include <stdint.h>

// Problem constants (from reference): B=4, L1=128, L2=256, D1=512, D2=512, D=1024
#define B_   4
#define L1_  128
#define L2_  256
#define D1_  512
#define D2_  512
#define D_   1024

typedef __attribute__((ext_vector_type(2))) float v2f;
typedef __attribute__((ext_vector_type(4))) float v4f;
typedef __attribute__((ext_vector_type(8))) float v8f;

// ---------------------------------------------------------------------------
// f32 WMMA: V_WMMA_F32_16X16X4_F32, D = A(16x4) * B(4x16) + C(16x16)
// ---------------------------------------------------------------------------
__device__ __forceinline__ v8f wmma_f32(v2f a, v2f b, v8f c) {
    return __builtin_amdgcn_wmma_f32_16x16x4_f32(
        /*neg_a=*/false, a, /*neg_b=*/false, b,
        /*c_mod=*/(short)0, c, /*reuse_a=*/false, /*reuse_b=*/false);
}

__device__ __forceinline__ float fast_tanh(float x) {
#if __has_builtin(__builtin_amdgcn_tanhf)
    return __builtin_amdgcn_tanhf(x);   // v_tanh_f32 on gfx1250
#else
    return tanhf(x);
#endif
}

// Direct memory -> LDS copy (16B per lane), tracked by ASYNCcnt.
__device__ __forceinline__ void async_lds_b128(uint32_t lds_byte_off, const float* gptr) {
    asm volatile("global_load_async_to_lds_b128 %0, %1, off"
                 :: "v"(lds_byte_off), "v"(gptr) : "memory");
}
__device__ __forceinline__ void wait_asynccnt0() {
#if __has_builtin(__builtin_amdgcn_s_wait_asynccnt)
    __builtin_amdgcn_s_wait_asynccnt(0);
#else
    asm volatile("s_wait_asynccnt 0" ::: "memory");
#endif
}

// ---------------------------------------------------------------------------
// Phase 1: ka = keys @ Wa^T (row tiles 0..31), vb = values @ Wb^T (32..95).
// One wave per 16x64 output strip; K stepped by 4; software-pipelined so
// next-step loads overlap the 4 WMMAs of the current step.
// A-layout (16x4 f32): lane m = lane&15, VGPR0/1 hold K = 2*(lane>>4)+{0,1}.
// B-layout (4x16 f32): lane n = lane&15, VGPR0/1 hold K = 2*(lane>>4)+{0,1}.
// Both operands are contiguous along K in memory -> 64-bit lane loads.
// ---------------------------------------------------------------------------
__global__ __launch_bounds__(32)
void proj_kernel(const float* __restrict__ keys,   // (B*L1, D1)
                 const float* __restrict__ values, // (B*L2, D2)
                 const float* __restrict__ W,      // (D, D)
                 float* __restrict__ ka,           // (B*L1, D)
                 float* __restrict__ vb)           // (B*L2, D)
{
    const int rt   = blockIdx.x;          // 0..95
    const int ct   = blockIdx.y;          // 0..15 -> 64 columns each
    const int lane = threadIdx.x;
    const int lm   = lane & 15;
    const int kb   = (lane >> 4) << 1;    // 0 or 2

    const float* X; float* Y; int kOff; int row0;
    if (rt < 32) { X = keys;   Y = ka; kOff = 0;   row0 = rt * 16; }
    else         { X = values; Y = vb; kOff = D1_; row0 = (rt - 32) * 16; }

    const int col0 = ct * 64;
    const float* xrow = X + (size_t)(row0 + lm) * D1_ + kb;
    const float* w0 = W + (size_t)(col0 + 0  + lm) * D_ + kOff + kb;
    const float* w1 = W + (size_t)(col0 + 16 + lm) * D_ + kOff + kb;
    const float* w2 = W + (size_t)(col0 + 32 + lm) * D_ + kOff + kb;
    const float* w3 = W + (size_t)(col0 + 48 + lm) * D_ + kOff + kb;

    v8f acc0 = {}, acc1 = {}, acc2 = {}, acc3 = {};

    // pipeline prologue: k = 0 fragments
    v2f a_c  = *(const v2f*)(xrow);
    v2f b0_c = *(const v2f*)(w0);
    v2f b1_c = *(const v2f*)(w1);
    v2f b2_c = *(const v2f*)(w2);
    v2f b3_c = *(const v2f*)(w3);

    #pragma unroll 4
    for (int k = 0; k < D1_ - 4; k += 4) {
        const int kn = k + 4;
        const v2f a_n  = *(const v2f*)(xrow + kn);
        const v2f b0_n = *(const v2f*)(w0 + kn);
        const v2f b1_n = *(const v2f*)(w1 + kn);
        const v2f b2_n = *(const v2f*)(w2 + kn);
        const v2f b3_n = *(const v2f*)(w3 + kn);
        acc0 = wmma_f32(a_c, b0_c, acc0);
        acc1 = wmma_f32(a_c, b1_c, acc1);
        acc2 = wmma_f32(a_c, b2_c, acc2);
        acc3 = wmma_f32(a_c, b3_c, acc3);
        a_c = a_n; b0_c = b0_n; b1_c = b1_n; b2_c = b2_n; b3_c = b3_n;
    }
    acc0 = wmma_f32(a_c, b0_c, acc0);
    acc1 = wmma_f32(a_c, b1_c, acc1);
    acc2 = wmma_f32(a_c, b2_c, acc2);
    acc3 = wmma_f32(a_c, b3_c, acc3);

    // D layout: VGPR v -> M = v + 8*(lane>>4), N = lane&15
    const int mhi = (lane >> 4) << 3;
    #pragma unroll
    for (int v = 0; v < 8; ++v) {
        float* yrow = Y + (size_t)(row0 + mhi + v) * D_ + col0 + lm;
        yrow[0]  = acc0[v];
        yrow[16] = acc1[v];
        yrow[32] = acc2[v];
        yrow[48] = acc3[v];
    }
}

// ---------------------------------------------------------------------------
// Phase 2: logits[b,i,j] = sum_d tanh(ka[b,i,d] + vb[b,j,d]) * V[d]
// 16x16 (i,j) tile per 256-thread workgroup; d chunked by 256 through LDS
// (16x reuse of every ka/vb row). Tiles are filled with CDNA5 async
// memory->LDS B128 copies (ASYNCcnt), inner loop reads ds_load_b128.
// Row pad of 4 floats: rows stay 16B-aligned and banks 4*tx+d are disjoint
// across the 16 tx lanes of a wave -> conflict-free.
// ---------------------------------------------------------------------------
#define DCH  256
#define LPAD 4
__global__ __launch_bounds__(256)
void logits_kernel(const float* __restrict__ ka,   // (B*L1, D)
                   const float* __restrict__ vb,   // (B*L2, D)
                   const float* __restrict__ V,    // (D)
                   float* __restrict__ logits)     // (B, L1, L2)
{
    __shared__ float kas[16][DCH + LPAD];
    __shared__ float vbs[16][DCH + LPAD];
    __shared__ float Vs[DCH];

    const int b   = blockIdx.z;            // 0..3
    const int i0  = blockIdx.y * 16;
    const int j0  = blockIdx.x * 16;
    const int tid = threadIdx.x;
    const int tx  = tid & 15;              // j local
    const int ty  = tid >> 4;              // i local

    const uint32_t kas_base = (uint32_t)(uintptr_t)&kas[0][0];
    const uint32_t vbs_base = (uint32_t)(uintptr_t)&vbs[0][0];
    const uint32_t vs_base  = (uint32_t)(uintptr_t)&Vs[0];

    const float* krow = &kas[ty][0];
    const float* vrow = &vbs[tx][0];

    float acc = 0.f;
    for (int d0 = 0; d0 < D_; d0 += DCH) {
        __syncthreads();  // previous chunk fully consumed before refill
        // 16 rows x 64 float4-groups = 1024 groups; 256 lanes -> 4 passes each
        #pragma unroll
        for (int p = 0; p < 4; ++p) {
            const int idx = p * 256 + tid;
            const int r   = idx >> 6;              // row 0..15
            const int c   = (idx & 63) << 2;       // col float (16B groups)
            const uint32_t loff = (uint32_t)((r * (DCH + LPAD) + c) << 2);
            async_lds_b128(kas_base + loff,
                           ka + (size_t)(b * L1_ + i0 + r) * D_ + d0 + c);
            async_lds_b128(vbs_base + loff,
                           vb + (size_t)(b * L2_ + j0 + r) * D_ + d0 + c);
        }
        if (tid < DCH / 4)
            async_lds_b128(vs_base + ((uint32_t)tid << 4), V + d0 + tid * 4);
        wait_asynccnt0();
        __syncthreads();  // all waves' async fills visible

        #pragma unroll 4
        for (int d = 0; d < DCH; d += 4) {
            const v4f kv = *(const v4f*)(krow + d);
            const v4f vv = *(const v4f*)(vrow + d);
            const v4f vs = *(const v4f*)(Vs + d);
            acc = fmaf(fast_tanh(kv.x + vv.x), vs.x, acc);
            acc = fmaf(fast_tanh(kv.y + vv.y), vs.y, acc);
            acc = fmaf(fast_tanh(kv.z + vv.z), vs.z, acc);
            acc = fmaf(fast_tanh(kv.w + vv.w), vs.w, acc);
        }
    }
    logits[((size_t)b * L1_ + i0 + ty) * L2_ + j0 + tx] = acc;
}

// ---------------------------------------------------------------------------
// Phase 3: softmax over j (256) per (b,i) row; in-place logits -> alpha
// ---------------------------------------------------------------------------
__global__ __launch_bounds__(256)
void softmax_kernel(float* __restrict__ logits)    // (B*L1, L2)
{
    __shared__ float red[256];
    const int tid = threadIdx.x;
    float* p = logits + (size_t)blockIdx.x * L2_;
    const float x = p[tid];

    red[tid] = x; __syncthreads();
    for (int s = 128; s > 0; s >>= 1) {
        if (tid < s) red[tid] = fmaxf(red[tid], red[tid + s]);
        __syncthreads();
    }
    const float m = red[0];
    __syncthreads();
    const float e = __expf(x - m);
    red[tid] = e; __syncthreads();
    for (int s = 128; s > 0; s >>= 1) {
        if (tid < s) red[tid] += red[tid + s];
        __syncthreads();
    }
    p[tid] = e / red[0];
}

// ---------------------------------------------------------------------------
// Phase 4: out[b] = alpha[b] (128x256) @ values[b] (256x512), f32 WMMA,
// software-pipelined like phase 1.
// ---------------------------------------------------------------------------
__global__ __launch_bounds__(32)
void out_kernel(const float* __restrict__ alpha,   // (B, L1, L2)
                const float* __restrict__ values,  // (B, L2, D2)
                float* __restrict__ out)           // (B, L1, D2)
{
    const int b    = blockIdx.z;
    const int mt   = blockIdx.y;          // 0..7
    const int nt   = blockIdx.x;          // 0..7 -> 64 cols
    const int lane = threadIdx.x;
    const int lm   = lane & 15;
    const int kb   = (lane >> 4) << 1;

    const float* Arow = alpha + ((size_t)b * L1_ + mt * 16 + lm) * L2_ + kb;
    const float* Vals = values + (size_t)b * L2_ * D2_;
    const int n0 = nt * 64;

    v8f acc0 = {}, acc1 = {}, acc2 = {}, acc3 = {};

    // B fragment for step k: rows k+kb, k+kb+1 of values at column n
    const float* r0 = Vals + (size_t)kb * D2_ + n0 + lm;
    v2f a_c  = *(const v2f*)(Arow);
    v2f b0_c = { r0[0],  r0[D2_ + 0]  };
    v2f b1_c = { r0[16], r0[D2_ + 16] };
    v2f b2_c = { r0[32], r0[D2_ + 32] };
    v2f b3_c = { r0[48], r0[D2_ + 48] };

    #pragma unroll 4
    for (int k = 0; k < L2_ - 4; k += 4) {
        const float* rn = Vals + (size_t)(k + 4 + kb) * D2_ + n0 + lm;
        const v2f a_n  = *(const v2f*)(Arow + k + 4);
        const v2f b0_n = { rn[0],  rn[D2_ + 0]  };
        const v2f b1_n = { rn[16], rn[D2_ + 16] };
        const v2f b2_n = { rn[32], rn[D2_ + 32] };
        const v2f b3_n = { rn[48], rn[D2_ + 48] };
        acc0 = wmma_f32(a_c, b0_c, acc0);
        acc1 = wmma_f32(a_c, b1_c, acc1);
        acc2 = wmma_f32(a_c, b2_c, acc2);
        acc3 = wmma_f32(a_c, b3_c, acc3);
        a_c = a_n; b0_c = b0_n; b1_c = b1_n; b2_c = b2_n; b3_c = b3_n;
    }
    acc0 = wmma_f32(a_c, b0_c, acc0);
    acc1 = wmma_f32(a_c, b1_c, acc1);
    acc2 = wmma_f32(a_c, b2_c, acc2);
    acc3 = wmma_f32(a_c, b3_c, acc3);

    const int mhi = (lane >> 4) << 3;
    #pragma unroll
    for (int v = 0; v < 8; ++v) {
        float* orow = out + ((size_t)b * L1_ + mt * 16 + mhi + v) * D2_ + n0 + lm;
        orow[0]  = acc0[v];
        orow[16] = acc1[v];
        orow[32] = acc2[v];
        orow[48] = acc3[v];
    }
}

// ---------------------------------------------------------------------------
extern "C" void kernel_launch(void* const* d_in, const int* in_sizes, int n_in,
                              void* d_out, int out_size, void* d_ws, size_t ws_size,
                              hipStream_t stream) {
    const float* keys   = (const float*)d_in[0];   // (4,128,512)
    const float* values = (const float*)d_in[1];   // (4,256,512)
    const float* W      = (const float*)d_in[2];   // (1024,1024)
    const float* V      = (const float*)d_in[3];   // (1,1024)
    float* out = (float*)d_out;                    // (4,128,512)

    // Workspace layout: ka (2 MB) | vb (4 MB) | logits/alpha (512 KB)
    float* ka     = (float*)d_ws;
    float* vb     = ka + (size_t)B_ * L1_ * D_;
    float* logits = vb + (size_t)B_ * L2_ * D_;

    proj_kernel<<<dim3(96, 16), 32, 0, stream>>>(keys, values, W, ka, vb);
    logits_kernel<<<dim3(L2_ / 16, L1_ / 16, B_), 256, 0, stream>>>(ka, vb, V, logits);
    softmax_kernel<<<B_ * L1_, 256, 0, stream>>>(logits);
    out_kernel<<<dim3(D2_ / 64, L1_ / 16, B_), 32, 0, stream>>>(logits, values, out);
}